// Moment_31241592111833
// MI455X (gfx1250) — compile-verified
//
#include <hip/hip_runtime.h>
#include <cstdint>

// ---------------- problem constants ----------------
#define NB      16384      // batch rows
#define ND      32         // latent dim
#define NFEAT   6545       // 1 + 32 + 528 + 5984 features per row
#define NPAD    6548       // table padded to multiple of 4 dwords
#define THREADS 1024
#define DW_PER_BLOCK 16384u   // output dwords produced per block (16 per thread)

typedef unsigned int v4u __attribute__((ext_vector_type(4)));
typedef int          v8i __attribute__((ext_vector_type(8)));
typedef int          v4i __attribute__((ext_vector_type(4)));
typedef float        v4f __attribute__((ext_vector_type(4)));

// ---------------- compile-time (i,k,j) monomial index table ----------------
// Entry packs three channel indices (8 bits each); index 32 is a sentinel
// channel holding 1.0f, so every feature is uniformly xr[a]*xr[b]*xr[c].
// Ordering mirrors the reference exactly:
//   col 0          : ones
//   cols 1..32     : x_j                       (j = 0..31)
//   next 528       : for i: for j>=i:  x_i x_j
//   next 5984      : for i: for k>=i: for j>=k: x_i x_k x_j
struct Tab { unsigned v[NPAD]; };

static constexpr unsigned pack3(unsigned a, unsigned b, unsigned c) {
    return a | (b << 8) | (c << 16);
}

static constexpr Tab make_table() {
    Tab t{};
    int c = 0;
    t.v[c++] = pack3(32u, 32u, 32u);
    for (unsigned j = 0; j < ND; ++j) t.v[c++] = pack3(j, 32u, 32u);
    for (unsigned i = 0; i < ND; ++i)
        for (unsigned j = i; j < ND; ++j) t.v[c++] = pack3(i, j, 32u);
    for (unsigned i = 0; i < ND; ++i)
        for (unsigned k = i; k < ND; ++k)
            for (unsigned j = k; j < ND; ++j) t.v[c++] = pack3(i, k, j);
    for (; c < NPAD; ++c) t.v[c] = pack3(32u, 32u, 32u);
    return t;
}

__device__ constexpr Tab g_tab = make_table();

// ---------------- kernel ----------------
__global__ __launch_bounds__(THREADS)
void moment_features_kernel(const float* __restrict__ x,
                            float* __restrict__ out,
                            unsigned n_dwords) {
    __shared__ unsigned lds_tab[NPAD];   // 26192 B
    __shared__ float    xl[4 * 33];      // up to 4 rows of x, +1.0 sentinel

    const unsigned tid   = threadIdx.x;
    const unsigned blk   = blockIdx.x;
    const unsigned flat0 = blk * DW_PER_BLOCK;
    const unsigned r0    = flat0 / (unsigned)NFEAT;   // first row this block touches

    // ---- TDM: async-DMA the 26 KB index table into LDS (wave 0 issues it) ----
    if (tid < 32u) {
        unsigned long long ga = (unsigned long long)(uintptr_t)&g_tab.v[0];
        unsigned lds_off = (unsigned)(uintptr_t)lds_tab;  // low 32 bits of flat = LDS offset

        v4u g0;
        g0.x = 1u;                                         // count = 1 (valid descriptor)
        g0.y = lds_off;                                    // lds_addr (bytes)
        g0.z = (unsigned)(ga & 0xffffffffull);             // global_addr[31:0]
        g0.w = (unsigned)((ga >> 32) & 0x1ffffffull)       // global_addr[56:32]
             | (2u << 30);                                 // type = 2 ("image")

        v8i g1;
        g1[0] = (int)(2u << 16);                           // data_size = 2 -> 4 bytes
        g1[1] = (int)(((unsigned)NPAD & 0xffffu) << 16);   // tensor_dim0[15:0]
        g1[2] = (int)(((unsigned)NPAD >> 16) | (1u << 16));// tensor_dim0[31:16] | tensor_dim1=1
        g1[3] = (int)(((unsigned)NPAD & 0xffffu) << 16);   // tile_dim0 = NPAD
        g1[4] = 1;                                         // tile_dim1 = 1, tile_dim2 = 0
        g1[5] = (int)NPAD;                                 // tensor_dim0_stride (low 32)
        g1[6] = 0;
        g1[7] = 0;

        v4i z4 = {0, 0, 0, 0};
        v8i z8 = {0, 0, 0, 0, 0, 0, 0, 0};
        __builtin_amdgcn_tensor_load_to_lds(g0, g1, z4, z4, z8, 0);
        __builtin_amdgcn_s_wait_tensorcnt(0);
    }

    // ---- stage <=4 rows of x into LDS; channel 32 = 1.0 sentinel ----
    if (tid < 4u * 33u) {
        unsigned rl = tid / 33u;
        unsigned ch = tid - rl * 33u;
        unsigned r  = r0 + rl;
        if (r > (unsigned)(NB - 1)) r = (unsigned)(NB - 1);
        xl[tid] = (ch < (unsigned)ND) ? x[r * ND + ch] : 1.0f;
    }
    __syncthreads();
    asm volatile("" ::: "memory");   // keep LDS reads below the barrier

    // ---- produce 4 float4 chunks per thread, coalesced & 16B-aligned ----
    const unsigned cbase = blk * (DW_PER_BLOCK / 4u);
    #pragma unroll
    for (int it = 0; it < 4; ++it) {
        unsigned c    = cbase + (unsigned)it * THREADS + tid;
        unsigned flat = c * 4u;
        if (flat >= n_dwords) continue;

        unsigned row = flat / (unsigned)NFEAT;            // magic-number division
        unsigned col = flat - row * (unsigned)NFEAT;

        float v[4];
        #pragma unroll
        for (int e = 0; e < 4; ++e) {
            unsigned ce = col + (unsigned)e;
            unsigned re = row;
            if (ce >= (unsigned)NFEAT) { ce -= (unsigned)NFEAT; re += 1u; }
            unsigned ent = lds_tab[ce];
            const float* xr = &xl[(re - r0) * 33u];
            v[e] = xr[ent & 255u] * xr[(ent >> 8) & 255u] * xr[(ent >> 16) & 255u];
        }

        v4f o = { v[0], v[1], v[2], v[3] };
        __builtin_nontemporal_store(o, (v4f*)(out + flat));   // NT B128 store
    }
}

// ---------------- launcher ----------------
extern "C" void kernel_launch(void* const* d_in, const int* in_sizes, int n_in,
                              void* d_out, int out_size, void* d_ws, size_t ws_size,
                              hipStream_t stream) {
    (void)in_sizes; (void)n_in; (void)d_ws; (void)ws_size;
    const float* x  = (const float*)d_in[0];
    float* out      = (float*)d_out;

    unsigned n_dwords = (unsigned)out_size;               // 16384 * 6545 = 107,233,280
    unsigned blocks   = (n_dwords + DW_PER_BLOCK - 1u) / DW_PER_BLOCK;  // 6545

    moment_features_kernel<<<dim3(blocks), dim3(THREADS), 0, stream>>>(x, out, n_dwords);
}